// LearnableSeriesDecomp_64991445123791
// MI455X (gfx1250) — compile-verified
//
#include <hip/hip_runtime.h>
#include <math.h>

// Problem constants (from reference)
#define B_    32
#define L_    4096
#define D_    256
#define K_    50
#define HID_  128
#define MIN_K 5
#define NCH   16          // L-chunks for two-stage mean reduction
#define TROWS 4           // t-rows per thread (register sliding window)

typedef __attribute__((ext_vector_type(2))) float v2f;
typedef __attribute__((ext_vector_type(8))) float v8f;

// Workspace layout (float offsets)
#define WS_XG    0                       // [B_*D_]      x_global
#define WS_W     (WS_XG + B_*D_)         // [B_*K_]      per-sample tap weights
#define WS_INT   (WS_W + B_*K_)          // [2*B_] ints  half[B_], k[B_]
#define WS_PART  (WS_INT + 2*B_)         // [B_*NCH*D_]  partial sums

// ---------------------------------------------------------------------------
// Stage 1a: partial sums over L chunks.  grid=(B,NCH), block=D (256 threads)
// ---------------------------------------------------------------------------
__global__ void mean_partial_kernel(const float* __restrict__ x,
                                    float* __restrict__ part) {
    const int b  = blockIdx.x;
    const int ch = blockIdx.y;
    const int d  = threadIdx.x;
    const int tpc = L_ / NCH;            // 256 timesteps per chunk
    float acc = 0.0f;
    size_t base = ((size_t)b * L_ + (size_t)ch * tpc) * D_ + d;
    for (int t = 0; t < tpc; ++t)
        acc += x[base + (size_t)t * D_];
    part[((size_t)b * NCH + ch) * D_ + d] = acc;
}

// Stage 1b: reduce NCH partials -> mean.  grid=B, block=D
__global__ void mean_final_kernel(const float* __restrict__ part,
                                  float* __restrict__ xg) {
    const int b = blockIdx.x;
    const int d = threadIdx.x;
    float acc = 0.0f;
    for (int c = 0; c < NCH; ++c)
        acc += part[((size_t)b * NCH + c) * D_ + d];
    xg[b * D_ + d] = acc * (1.0f / (float)L_);
}

// ---------------------------------------------------------------------------
// Stage 2: MLP via f32 WMMA.  One workgroup, 512 threads = 16 wave32 waves.
// GEMM: h[32,128] = relu(xg[32,256] @ W1[256,128] + b1)
// Each wave owns one 16x16 output tile, iterating K in steps of 4 with
// V_WMMA_F32_16X16X4_F32.  Fragment layouts per ISA 7.12.2:
//   A (16x4, MxK):  VGPR i, lanes 0-15 -> K=i, lanes 16-31 -> K=i+2; M=lane&15
//   B (4x16, KxN):  VGPR i, lanes 0-15 -> K=i, lanes 16-31 -> K=i+2; N=lane&15
//   C/D (16x16):    VGPR i, lanes 0-15 -> M=i, lanes 16-31 -> M=i+8; N=lane&15
// Then lanes 0..31 finish: logit = sigmoid(h@W2 + b2), k, half, masked softmax.
// ---------------------------------------------------------------------------
__global__ void __launch_bounds__(512)
mlp_wmma_kernel(const float* __restrict__ xg,
                const float* __restrict__ tw,
                const float* __restrict__ W1,
                const float* __restrict__ b1,
                const float* __restrict__ W2,
                const float* __restrict__ b2,
                float* __restrict__ wOut,
                int*   __restrict__ halfOut,
                int*   __restrict__ kOut) {
    __shared__ float h_lds[B_ * HID_];   // 16 KB

    const int tid  = threadIdx.x;
    const int lane = tid & 31;
    const int wave = tid >> 5;           // 0..15
    const int m0   = (wave >> 3) * 16;   // 0 or 16
    const int n0   = (wave & 7) * 16;    // 0..112
    const int lh   = lane >> 4;          // 0 | 1 (lane half)
    const int ll   = lane & 15;

    v8f c = {};
    for (int kk = 0; kk < D_; kk += 4) {
        v2f a, bm;
        // A fragment: rows m0+ll of xg, K columns kk..kk+3
        a.x = xg[(m0 + ll) * D_ + kk + (lh ? 2 : 0)];
        a.y = xg[(m0 + ll) * D_ + kk + (lh ? 3 : 1)];
        // B fragment: K rows kk..kk+3 of W1, columns n0+ll
        bm.x = W1[(kk + (lh ? 2 : 0)) * HID_ + n0 + ll];
        bm.y = W1[(kk + (lh ? 3 : 1)) * HID_ + n0 + ll];
        c = __builtin_amdgcn_wmma_f32_16x16x4_f32(
                /*neg_a=*/false, a, /*neg_b=*/false, bm,
                /*c_mod=*/(short)0, c, /*reuse_a=*/false, /*reuse_b=*/false);
    }

    // bias + relu, scatter C/D fragment into LDS
    const float bias = b1[n0 + ll];
#pragma unroll
    for (int i = 0; i < 8; ++i) {
        const int row = m0 + i + lh * 8;
        float v = c[i] + bias;
        h_lds[row * HID_ + n0 + ll] = v > 0.0f ? v : 0.0f;
    }
    __syncthreads();

    // Head: one lane per sample b
    if (tid < B_) {
        const int b = tid;
        float dot = 0.0f;
        for (int j = 0; j < HID_; ++j)
            dot += h_lds[b * HID_ + j] * W2[j];
        const float logit = 1.0f / (1.0f + __expf(-(dot + b2[0])));
        int k = (int)roundf(logit * (float)(K_ - MIN_K) + (float)MIN_K);
        const int kmax = (K_ < (L_ / 3)) ? K_ : (L_ / 3);   // = 50
        if (k < MIN_K) k = MIN_K;
        if (k > kmax)  k = kmax;
        const int half = k >> 1;

        // masked softmax over first k taps of trend_weights
        float m = -3.402823466e38f;
        for (int j = 0; j < k; ++j) m = fmaxf(m, tw[j]);
        float s = 0.0f;
        for (int j = 0; j < k; ++j) s += __expf(tw[j] - m);
        const float inv = 1.0f / s;
        for (int j = 0; j < K_; ++j)
            wOut[b * K_ + j] = (j < k) ? __expf(tw[j] - m) * inv : 0.0f;
        halfOut[b] = half;
        kOut[b]    = k;
    }
}

// ---------------------------------------------------------------------------
// Stage 3: fused trend + seasonal with register-blocked sliding window.
// grid=(L/16, B), block=256 (64 float4 d-columns x 4 t-groups).
// Each thread owns TROWS=4 consecutive t-rows of one float4 d-slice: the
// union window is k+3 rows, so each global_load_b128 feeds 4 accumulators
// (~3.8x fewer loads than the naive form).  Tap weights are staged in LDS
// zero-padded by TROWS-1 on the left so the shifted lookup is branch-free:
//   acc[r] += swp[(TROWS-1) + jj - r] * x[clamp(t + jj - half)]
// (taps >= k are already zero from the masked softmax).
// ---------------------------------------------------------------------------
__global__ void __launch_bounds__(256)
decomp_kernel(const float* __restrict__ x,
              const float* __restrict__ w,
              const int* __restrict__ halfArr,
              const int* __restrict__ kArr,
              float* __restrict__ seasonal,
              float* __restrict__ trend) {
    __shared__ float swp[K_ + 2 * TROWS];   // padded tap table
    __shared__ int   sh, skk;

    const int b   = blockIdx.y;
    const int t0  = blockIdx.x * (4 * TROWS);   // 16 t-rows per block
    const int tid = threadIdx.x;

    if (tid < K_ + 2 * TROWS) {
        const int i = tid - (TROWS - 1);        // tap index
        swp[tid] = (i >= 0 && i < K_) ? w[b * K_ + i] : 0.0f;
    }
    if (tid == 0) { sh = halfArr[b]; skk = kArr[b]; }
    __syncthreads();

    const int d4 = tid & 63;                    // float4 column
    const int t  = t0 + (tid >> 6) * TROWS;     // first of TROWS rows
    const int hf = sh;
    const int kk = skk;

    const float4* __restrict__ x4 = (const float4*)x;
    const size_t rowBase = (size_t)b * L_ * (D_ / 4);

    // Prefetch the far edge of the union window (global_prefetch_b8)
    {
        int pidx = t + kk + TROWS - 2 - hf;
        pidx = pidx < 0 ? 0 : (pidx > L_ - 1 ? L_ - 1 : pidx);
        __builtin_prefetch(&x4[rowBase + (size_t)pidx * (D_ / 4) + d4], 0, 1);
    }

    float4 acc[TROWS];
#pragma unroll
    for (int r = 0; r < TROWS; ++r) { acc[r].x = acc[r].y = acc[r].z = acc[r].w = 0.0f; }

    const int jEnd = kk + TROWS - 1;            // union window length
    for (int jj = 0; jj < jEnd; ++jj) {
        int idx = t + jj - hf;
        idx = idx < 0 ? 0 : (idx > L_ - 1 ? L_ - 1 : idx);
        const float4 v = x4[rowBase + (size_t)idx * (D_ / 4) + d4];
#pragma unroll
        for (int r = 0; r < TROWS; ++r) {
            const float wj = swp[(TROWS - 1) + jj - r];
            acc[r].x = fmaf(wj, v.x, acc[r].x);
            acc[r].y = fmaf(wj, v.y, acc[r].y);
            acc[r].z = fmaf(wj, v.z, acc[r].z);
            acc[r].w = fmaf(wj, v.w, acc[r].w);
        }
    }

#pragma unroll
    for (int r = 0; r < TROWS; ++r) {
        const size_t o = rowBase + (size_t)(t + r) * (D_ / 4) + d4;
        const float4 xv = x4[o];                // L0-hot: inside the window
        float4 se;
        se.x = xv.x - acc[r].x; se.y = xv.y - acc[r].y;
        se.z = xv.z - acc[r].z; se.w = xv.w - acc[r].w;
        ((float4*)seasonal)[o] = se;
        ((float4*)trend)[o]    = acc[r];
    }
}

// ---------------------------------------------------------------------------
extern "C" void kernel_launch(void* const* d_in, const int* in_sizes, int n_in,
                              void* d_out, int out_size, void* d_ws, size_t ws_size,
                              hipStream_t stream) {
    const float* x  = (const float*)d_in[0];
    const float* tw = (const float*)d_in[1];
    const float* W1 = (const float*)d_in[2];
    const float* b1 = (const float*)d_in[3];
    const float* W2 = (const float*)d_in[4];
    const float* b2 = (const float*)d_in[5];

    float* out      = (float*)d_out;
    float* seasonal = out;
    float* trend    = out + (size_t)B_ * L_ * D_;

    float* ws      = (float*)d_ws;
    float* xg      = ws + WS_XG;
    float* wBuf    = ws + WS_W;
    int*   ints    = (int*)(ws + WS_INT);
    int*   halfArr = ints;
    int*   kArr    = ints + B_;
    float* part    = ws + WS_PART;

    mean_partial_kernel<<<dim3(B_, NCH), D_, 0, stream>>>(x, part);
    mean_final_kernel<<<B_, D_, 0, stream>>>(part, xg);
    mlp_wmma_kernel<<<1, 512, 0, stream>>>(xg, tw, W1, b1, W2, b2,
                                           wBuf, halfArr, kArr);
    decomp_kernel<<<dim3(L_ / (4 * TROWS), B_), 256, 0, stream>>>(
        x, wBuf, halfArr, kArr, seasonal, trend);
}